// NormalLoss_51857435132223
// MI455X (gfx1250) — compile-verified
//
#include <hip/hip_runtime.h>

typedef __attribute__((ext_vector_type(2))) float v2f;
typedef __attribute__((ext_vector_type(4))) float v4f;
typedef __attribute__((ext_vector_type(8))) float v8f;

#define NPTS 4096
#define NN 10

// One workgroup = 256 threads = 8 wave32s; each wave owns 16 query columns.
// LDS holds interleaved {x,y,z,|p|^2} per point (64KB). The WMMA computes the
// selection score directly:  A[m] = {x,y,z,sq}_m,  B[.][n] = {-2x,-2y,-2z,1}_n
//   =>  D[m][n] = sq_m - 2*p_m.p_n   (= d2[m][n] - sq_n, same top-k ordering).
// Tile loop is 2x unrolled + software pipelined: two independent WMMAs issue
// back-to-back, next A-pair ds_load_b64s issue before the insert phase so the
// s_wait_dscnt at the next WMMA is already satisfied.
__global__ __launch_bounds__(256) void knn_normals_kernel(
    const float* __restrict__ pred, const float* __restrict__ gt,
    float* __restrict__ normals /* [2][8][3][4096] */)
{
    __shared__ v4f pts[NPTS];   // {x, y, z, sq} per point : 65536 bytes

    const int b = blockIdx.y;
    const float* src = (blockIdx.z == 0) ? pred : gt;
    const float* px = src + b * 3 * NPTS;
    const float* py = px + NPTS;
    const float* pz = px + 2 * NPTS;

    // Cooperative stage of the whole batch into LDS
    for (int i = threadIdx.x; i < NPTS; i += 256) {
        const float x = px[i], y = py[i], z = pz[i];
        v4f p; p.x = x; p.y = y; p.z = z; p.w = x * x + y * y + z * z;
        pts[i] = p;
    }
    __syncthreads();

    const int lane = threadIdx.x & 31;
    const int wave = threadIdx.x >> 5;
    const int half = lane >> 4;   // 0 -> K=0,1 / rows 0-7 ; 1 -> K=2,3 / rows 8-15
    const int ln   = lane & 15;
    const int q    = blockIdx.x * 128 + wave * 16 + ln;  // this lane's query column

    // B operand (4x16, K-major): lanes 0-15 hold K=0,1; lanes 16-31 hold K=2,3.
    // Column n = {-2x, -2y, -2z, 1} of query n.
    v2f bmat = ((const v2f*)&pts[q])[half];   // {x,y} or {z,sq}
    bmat.x *= -2.0f;
    bmat.y = half ? 1.0f : -2.0f * bmat.y;

    float bd[NN]; int bi[NN];
#pragma unroll
    for (int k = 0; k < NN; ++k) { bd[k] = 3.0e38f; bi[k] = 0; }

    // A operand (16x4): single unconditional b64 load from interleaved LDS.
    // lane<16: {x,y} of point t+ln (K=0,1); lane>=16: {z,sq} (K=2,3).
    auto loadA = [&](int t) -> v2f { return ((const v2f*)&pts[t + ln])[half]; };

    // Sorted top-10 insert (max at [NN-1], bubble-down on replace)
    auto insert8 = [&](const v8f& c, int mbase) {
#pragma unroll
        for (int r = 0; r < 8; ++r) {
            const float s = c[r];                 // sq_m - 2*dot  (score)
            if (s < bd[NN - 1]) {
                bd[NN - 1] = s; bi[NN - 1] = mbase + r;
#pragma unroll
                for (int k = NN - 1; k >= 1; --k) {
                    if (bd[k] < bd[k - 1]) {
                        float td = bd[k]; bd[k] = bd[k - 1]; bd[k - 1] = td;
                        int   ti = bi[k]; bi[k] = bi[k - 1]; bi[k - 1] = ti;
                    }
                }
            }
        }
    };

    v2f a0 = loadA(0);
    v2f a1 = loadA(16);
    for (int tile = 0; tile < NPTS; tile += 32) {
        v8f c0 = {};
        c0 = __builtin_amdgcn_wmma_f32_16x16x4_f32(
                false, a0, false, bmat, (short)0, c0, false, false);
        v8f c1 = {};
        c1 = __builtin_amdgcn_wmma_f32_16x16x4_f32(
                false, a1, false, bmat, (short)0, c1, false, false);

        // Branchless prefetch of the next A pair (wraps to tile 0 at the end,
        // result discarded) -- DS latency hidden behind the insert phase.
        const int nt = (tile + 32) & (NPTS - 1);
        a0 = loadA(nt);
        a1 = loadA(nt + 16);

        insert8(c0, tile + half * 8);
        insert8(c1, tile + 16 + half * 8);
    }

    // Merge lane n (rows 0-7 mod 16) with lane n+16 (rows 8-15): same score
    // offset (-sq_n) on both halves, so ordering is consistent.
    float md[2 * NN]; int mi[2 * NN];
    const int partner = lane ^ 16;
#pragma unroll
    for (int k = 0; k < NN; ++k) {
        md[k] = bd[k]; mi[k] = bi[k];
        md[NN + k] = __shfl(bd[k], partner, 32);
        mi[NN + k] = __shfl(bi[k], partner, 32);
    }

    if (half == 0) {
        // Select the 10 smallest of 20 (set only; mean/cov are order-invariant)
#pragma unroll
        for (int p = 0; p < NN; ++p) {
#pragma unroll
            for (int j = 2 * NN - 1; j > p; --j) {
                if (md[j] < md[j - 1]) {
                    float td = md[j]; md[j] = md[j - 1]; md[j - 1] = td;
                    int   ti = mi[j]; mi[j] = mi[j - 1]; mi[j - 1] = ti;
                }
            }
        }

        // Neighborhood mean + 3x3 covariance (divide by NN, matching reference)
        float gx[NN], gy[NN], gz[NN];
        float mx = 0.f, my = 0.f, mz = 0.f;
#pragma unroll
        for (int k = 0; k < NN; ++k) {
            const v4f p = pts[mi[k]];             // one ds_load_b128 per neighbor
            gx[k] = p.x; gy[k] = p.y; gz[k] = p.z;
            mx += p.x; my += p.y; mz += p.z;
        }
        const float inv = 1.0f / (float)NN;
        mx *= inv; my *= inv; mz *= inv;
        float cxx = 0.f, cxy = 0.f, cxz = 0.f, cyy = 0.f, cyz = 0.f, czz = 0.f;
#pragma unroll
        for (int k = 0; k < NN; ++k) {
            const float dx = gx[k] - mx, dy = gy[k] - my, dz = gz[k] - mz;
            cxx += dx * dx; cxy += dx * dy; cxz += dx * dz;
            cyy += dy * dy; cyz += dy * dz; czz += dz * dz;
        }
        cxx *= inv; cxy *= inv; cxz *= inv; cyy *= inv; cyz *= inv; czz *= inv;

        // Smallest eigenvalue of symmetric 3x3 (trigonometric / Smith method)
        float nx = 1.f, ny = 0.f, nz = 0.f;
        const float qm  = (cxx + cyy + czz) * (1.0f / 3.0f);
        const float axx = cxx - qm, ayy = cyy - qm, azz = czz - qm;
        const float p2  = axx * axx + ayy * ayy + azz * azz
                        + 2.0f * (cxy * cxy + cxz * cxz + cyz * cyz);
        if (p2 > 1e-24f) {
            const float p  = sqrtf(p2 * (1.0f / 6.0f));
            const float ip = 1.0f / p;
            const float bxx = axx * ip, byy = ayy * ip, bzz = azz * ip;
            const float bxy = cxy * ip, bxz = cxz * ip, byz = cyz * ip;
            float r = 0.5f * (bxx * (byy * bzz - byz * byz)
                            - bxy * (bxy * bzz - byz * bxz)
                            + bxz * (bxy * byz - byy * bxz));
            r = fminf(1.0f, fmaxf(-1.0f, r));
            const float phi  = acosf(r) * (1.0f / 3.0f);
            const float lmin = qm + 2.0f * p * cosf(phi + 2.0943951023931953f); // +2pi/3

            // Eigenvector: largest cross product among rows of (C - lmin*I)
            const float r0x = cxx - lmin, r0y = cxy,        r0z = cxz;
            const float r1x = cxy,        r1y = cyy - lmin, r1z = cyz;
            const float r2x = cxz,        r2y = cyz,        r2z = czz - lmin;
            const float v0x = r0y * r1z - r0z * r1y, v0y = r0z * r1x - r0x * r1z, v0z = r0x * r1y - r0y * r1x;
            const float v1x = r0y * r2z - r0z * r2y, v1y = r0z * r2x - r0x * r2z, v1z = r0x * r2y - r0y * r2x;
            const float v2x = r1y * r2z - r1z * r2y, v2y = r1z * r2x - r1x * r2z, v2z = r1x * r2y - r1y * r2x;
            const float n0 = v0x * v0x + v0y * v0y + v0z * v0z;
            const float n1 = v1x * v1x + v1y * v1y + v1z * v1z;
            const float n2 = v2x * v2x + v2y * v2y + v2z * v2z;
            float bx = v0x, by = v0y, bz = v0z, bn = n0;
            if (n1 > bn) { bx = v1x; by = v1y; bz = v1z; bn = n1; }
            if (n2 > bn) { bx = v2x; by = v2y; bz = v2z; bn = n2; }
            if (bn > 1e-30f) {
                const float rn = rsqrtf(bn);
                nx = bx * rn; ny = by * rn; nz = bz * rn;
            }
        }
        // Deterministic sign convention: largest-|component| positive
        const float ax = fabsf(nx), ay = fabsf(ny), az = fabsf(nz);
        const float key = (ax >= ay && ax >= az) ? nx : ((ay >= az) ? ny : nz);
        if (key < 0.0f) { nx = -nx; ny = -ny; nz = -nz; }

        float* outb = normals + (size_t)(blockIdx.z * 8 + b) * 3 * NPTS;
        outb[0 * NPTS + q] = nx;
        outb[1 * NPTS + q] = ny;
        outb[2 * NPTS + q] = nz;
    }
}

// Single-block deterministic MSE reduction over the 2 x (8*3*4096) normal buffers.
__global__ __launch_bounds__(256) void mse_kernel(
    const float* __restrict__ a, const float* __restrict__ b,
    float* __restrict__ out, int n)
{
    __shared__ float red[256];
    float s = 0.0f;
    for (int i = threadIdx.x; i < n; i += 256) {
        const float d = a[i] - b[i];
        s += d * d;
    }
    red[threadIdx.x] = s;
    __syncthreads();
    for (int off = 128; off > 0; off >>= 1) {
        if ((int)threadIdx.x < off) red[threadIdx.x] += red[threadIdx.x + off];
        __syncthreads();
    }
    if (threadIdx.x == 0) out[0] = red[0] / (float)n;
}

extern "C" void kernel_launch(void* const* d_in, const int* in_sizes, int n_in,
                              void* d_out, int out_size, void* d_ws, size_t ws_size,
                              hipStream_t stream) {
    const float* pred = (const float*)d_in[0];
    const float* gt   = (const float*)d_in[1];
    float* normals = (float*)d_ws;            // 2 * 8*3*4096 floats = 786 KB scratch

    dim3 grid(32, 8, 2);                      // query-chunk x batch x {pred,gt}
    knn_normals_kernel<<<grid, 256, 0, stream>>>(pred, gt, normals);

    const int n = 8 * 3 * NPTS;
    mse_kernel<<<1, 256, 0, stream>>>(normals, normals + n, (float*)d_out, n);
}